// MeanAggregator_14826227106018
// MI455X (gfx1250) — compile-verified
//
#include <hip/hip_runtime.h>
#include <cstdint>

#define B_      4096
#define U_      50000
#define D_      300
#define TILE_M  16
#define SLOT_CAP 2048
#define CHUNK   64

typedef __attribute__((ext_vector_type(2))) float        v2f;
typedef __attribute__((ext_vector_type(8))) float        v8f;
typedef __attribute__((ext_vector_type(4))) unsigned int v4u;
typedef __attribute__((ext_vector_type(8))) int          v8i;
typedef __attribute__((ext_vector_type(4))) int          v4i;

#if defined(__gfx1250__) && __has_builtin(__builtin_amdgcn_tensor_load_to_lds) && __has_builtin(__builtin_amdgcn_s_wait_tensorcnt)
#define HAVE_TDM 1
#else
#define HAVE_TDM 0
#endif

__global__ __launch_bounds__(256) void mean_agg_kernel(
    const int* __restrict__ row_idx, const int* __restrict__ col_idx,
    const float* __restrict__ embed, float* __restrict__ out, int E)
{
  __shared__ float          Xl[CHUNK * D_];        // gathered neighbor rows (76.8 KB)
  __shared__ int            seg[TILE_M + 1];
  __shared__ int            slot_col[SLOT_CAP];
  __shared__ unsigned char  slot_row[SLOT_CAP];
  __shared__ unsigned char  keptf[SLOT_CAP];
  __shared__ float          wrow[TILE_M];
  __shared__ int            degr[TILE_M];
  __shared__ int            nslots_sh;

  const int tid  = threadIdx.x;
  const int lane = tid & 31;
  const int wid  = tid >> 5;
  const int m    = lane & 15;   // matrix row (A) or column (B/C), per lane
  const int h    = lane >> 4;   // lane half selects K pair / M+8
  const int r0   = blockIdx.x * TILE_M;

  // ---- phase 0: segment starts via binary search (row_idx sorted) ----
  if (tid <= TILE_M) {
    int target = r0 + tid;
    int lo = 0, hi = E;
    while (lo < hi) { int mid = (lo + hi) >> 1; if (row_idx[mid] < target) lo = mid + 1; else hi = mid; }
    seg[tid] = lo;
  }
  if (tid < TILE_M) degr[tid] = 0;
  __syncthreads();

  const int ts = seg[0];
  int cnt = seg[TILE_M] - ts;
  if (cnt > SLOT_CAP) cnt = SLOT_CAP;

  // ---- phase 1: duplicate flags (".set" semantics: first occurrence wins) ----
  for (int k = tid; k < cnt; k += (int)blockDim.x) {
    int i = ts + k;
    int r = row_idx[i] - r0;
    int c = col_idx[i];
    int s = seg[r];
    unsigned char kept = 1;
    for (int j = s; j < i; ++j) if (col_idx[j] == c) { kept = 0; break; }
    keptf[k] = kept;
  }
  __syncthreads();

  // ---- phase 2: ordered compaction (serial => deterministic FP order).
  //      Pad the last chunk: slot_row=0xFF (matches no row -> A weight 0),
  //      slot_col=0 (gathers row 0 harmlessly; 0-weight kills it). ----
  if (tid == 0) {
    int n = 0;
    for (int k = 0; k < cnt; ++k) {
      if (keptf[k]) {
        int i = ts + k;
        int r = row_idx[i] - r0;
        slot_row[n] = (unsigned char)r;
        slot_col[n] = col_idx[i];
        degr[r] += 1;
        ++n;
      }
    }
    nslots_sh = n;
    int npad = (n + CHUNK - 1) & ~(CHUNK - 1);
    for (int k = n; k < npad; ++k) { slot_row[k] = 0xFFu; slot_col[k] = 0; }
  }
  __syncthreads();
  if (tid < TILE_M) {
    int d = degr[tid];
    wrow[tid] = (d > 0) ? (1.0f / (float)d) : 0.0f;  // empty row -> exact 0 output
  }
  __syncthreads();

  const int   nslots = nslots_sh;
  const int   nch    = (nslots + CHUNK - 1) / CHUNK;
  const float wm     = wrow[m];

  v8f acc0 = {0.f,0.f,0.f,0.f,0.f,0.f,0.f,0.f};
  v8f acc1 = {0.f,0.f,0.f,0.f,0.f,0.f,0.f,0.f};
  v8f acc2 = {0.f,0.f,0.f,0.f,0.f,0.f,0.f,0.f};
  const int ct0 = wid, ct1 = wid + 8, ct2 = wid + 16;   // 19 col-tiles of 16 cover D=300
  const int c0 = ct0 * 16 + m;                          // < 128
  const int c1 = ct1 * 16 + m;                          // < 256
  const int c2 = ct2 * 16 + m;                          // may exceed 299 on tile 18
  const bool c2ok = (c2 < D_);

  for (int ch = 0; ch < nch; ++ch) {
    const int base = ch * CHUNK;
    __syncthreads();   // previous chunk's consumers are done with Xl

#if HAVE_TDM
    // ---- TDM gather: 4 descriptors x 16 row-indices -> contiguous LDS rows.
    //      Branch-free: slot_col is chunk-padded, so always 16 indices. ----
    if (tid == 0) {
      unsigned long long ga = (unsigned long long)(size_t)embed;
      unsigned ldsbase = (unsigned)(size_t)&Xl[0];     // low 32 bits of LDS aperture addr = offset
      #pragma unroll
      for (int dsc = 0; dsc < 4; ++dsc) {
        const int b16 = base + dsc * 16;
        int iw[8];
        #pragma unroll
        for (int d = 0; d < 8; ++d) {
          unsigned i0 = (unsigned)slot_col[b16 + 2*d];
          unsigned i1 = (unsigned)slot_col[b16 + 2*d + 1];
          iw[d] = (int)((i0 & 0xFFFFu) | (i1 << 16));
        }
        // D# group0: count=1 | gather_mode(16-bit idx); lds_addr; global_addr; type=2
        v4u g0 = { 0x80000001u,
                   ldsbase + (unsigned)(dsc * 16 * D_ * 4),
                   (unsigned)ga,
                   ((unsigned)(ga >> 32) & 0x01FFFFFFu) | 0x80000000u };
        // D# group1: data_size=4B; tensor_dim0=300; tensor_dim1=U; tile_dim0=300;
        //            tile_dim1=16 indices; tensor_dim0_stride=300
        v8i g1 = { (int)(2u << 16),
                   (int)((unsigned)D_ << 16),
                   (int)(((unsigned)U_ & 0xFFFFu) << 16),
                   (int)((unsigned)D_ << 16),
                   16,
                   (int)D_,
                   0, 0 };
        v4i g2 = { iw[0], iw[1], iw[2], iw[3] };   // row indices 0..7
        v4i g3 = { iw[4], iw[5], iw[6], iw[7] };   // row indices 8..15
        v8i gx = { 0, 0, 0, 0, 0, 0, 0, 0 };       // 6-arg form: extra group (unused)
        __builtin_amdgcn_tensor_load_to_lds(g0, g1, g2, g3, gx, 0);
      }
    }
    __builtin_amdgcn_s_wait_tensorcnt(0);
#else
    // fallback: cooperative float4 gather of the (padded) chunk
    for (int t = tid; t < CHUNK * (D_ / 4); t += (int)blockDim.x) {
      int s = t / (D_ / 4), e4 = t % (D_ / 4);
      ((float4*)(Xl + s * D_))[e4] =
          ((const float4*)(embed + (size_t)slot_col[base + s] * D_))[e4];
    }
#endif
    __syncthreads();

    // ---- accumulate: C(16x16) += A(16x4 weights) x B(4x16 gathered feats) ----
    // Guard-free: padded slots have slot_row==0xFF (A=0) against finite B data.
    #pragma unroll
    for (int kb = 0; kb < CHUNK / 4; ++kb) {
      const int k0 = kb * 4 + 2 * h;   // A layout: VGPR0 holds K=2h, VGPR1 K=2h+1
      const int k1 = k0 + 1;
      v2f a;
      a.x = (slot_row[base + k0] == m) ? wm : 0.f;
      a.y = (slot_row[base + k1] == m) ? wm : 0.f;

      v2f b;
      b.x = Xl[k0 * D_ + c0];
      b.y = Xl[k1 * D_ + c0];
      acc0 = __builtin_amdgcn_wmma_f32_16x16x4_f32(false, a, false, b, (short)0, acc0, false, false);

      b.x = Xl[k0 * D_ + c1];
      b.y = Xl[k1 * D_ + c1];
      acc1 = __builtin_amdgcn_wmma_f32_16x16x4_f32(false, a, false, b, (short)0, acc1, false, false);

      if (wid < 3) {                   // wave-uniform: EXEC stays all-ones inside
        v2f b2;
        b2.x = c2ok ? Xl[k0 * D_ + c2] : 0.f;
        b2.y = c2ok ? Xl[k1 * D_ + c2] : 0.f;
        acc2 = __builtin_amdgcn_wmma_f32_16x16x4_f32(false, a, false, b2, (short)0, acc2, false, false);
      }
    }
  }

  // ---- epilogue: C/D layout = VGPR p: lanes0-15 -> M=p, lanes16-31 -> M=p+8 ----
  {
    #pragma unroll
    for (int p = 0; p < 8; ++p) out[(size_t)(r0 + p + h * 8) * D_ + c0] = acc0[p];
  }
  {
    #pragma unroll
    for (int p = 0; p < 8; ++p) out[(size_t)(r0 + p + h * 8) * D_ + c1] = acc1[p];
  }
  if (wid < 3 && c2ok) {
    #pragma unroll
    for (int p = 0; p < 8; ++p) out[(size_t)(r0 + p + h * 8) * D_ + c2] = acc2[p];
  }
}

extern "C" void kernel_launch(void* const* d_in, const int* in_sizes, int n_in,
                              void* d_out, int out_size, void* d_ws, size_t ws_size,
                              hipStream_t stream) {
  const int*   row = (const int*)d_in[0];
  const int*   col = (const int*)d_in[1];
  const float* emb = (const float*)d_in[2];
  float*       out = (float*)d_out;
  const int E = in_sizes[0];
  (void)n_in; (void)out_size; (void)d_ws; (void)ws_size;
  mean_agg_kernel<<<dim3(B_ / TILE_M), dim3(256), 0, stream>>>(row, col, emb, out, E);
}